// TAN_83966610637209
// MI455X (gfx1250) — compile-verified
//
#include <hip/hip_runtime.h>
#include <hip/hip_bf16.h>
#include <cstddef>

// ---------------- constants ----------------
#define NSEG 64
#define DMODEL 256
#define HDIM 512
#define NVALID 1104
#define BATCH 4          // bs*ns = 2*2
#define NT 4             // N-tiles (16 cols each) per wave

typedef __attribute__((ext_vector_type(2))) float v2f;
typedef __attribute__((ext_vector_type(8))) float v8f;

// ---------------- compile-time mask tables ----------------
struct Tables {
  unsigned char vi[NVALID];
  unsigned char vj[NVALID];
  float mw0[66 * 66];
  float mw1[64 * 64];
};

constexpr Tables makeTables() {
  Tables t{};
  bool mask[64][64] = {};
  for (int i = 0; i < 64; ++i) mask[i][i] = true;
  {
    const int counts[3] = {15, 8, 8};
    int stride = 1, offset = 0;
    for (int g = 0; g < 3; ++g) {
      for (int c = 0; c < counts[g]; ++c) {
        offset += stride;
        for (int i = 0; i + offset < 64; i += stride) mask[i][i + offset] = true;
      }
      stride *= 2;
    }
  }
  int v = 0;
  for (int i = 0; i < 64; ++i)
    for (int j = 0; j < 64; ++j)
      if (mask[i][j]) { t.vi[v] = (unsigned char)i; t.vj[v] = (unsigned char)j; ++v; }
  for (int oy = 0; oy < 66; ++oy)
    for (int ox = 0; ox < 66; ++ox) {
      int cnt = 0;
      for (int ky = 0; ky < 3; ++ky)
        for (int kx = 0; kx < 3; ++kx) {
          int iy = oy + ky - 2, ix = ox + kx - 2;
          if (iy >= 0 && iy < 64 && ix >= 0 && ix < 64 && mask[iy][ix]) ++cnt;
        }
      t.mw0[oy * 66 + ox] = cnt > 0 ? 1.0f / (float)cnt : 0.0f;
    }
  for (int oy = 0; oy < 64; ++oy)
    for (int ox = 0; ox < 64; ++ox) {
      int cnt = 0;
      for (int ky = 0; ky < 3; ++ky)
        for (int kx = 0; kx < 3; ++kx)
          if (t.mw0[(oy + ky) * 66 + (ox + kx)] > 0.0f) ++cnt;
      t.mw1[oy * 64 + ox] = cnt > 0 ? 1.0f / (float)cnt : 0.0f;
    }
  return t;
}

__device__ constexpr Tables g_tab = makeTables();

// ---------------- utility kernels ----------------
__global__ void zero_kernel(float* p, long n) {
  long i = (long)blockIdx.x * blockDim.x + threadIdx.x;
  long stride = (long)gridDim.x * blockDim.x;
  for (; i < n; i += stride) p[i] = 0.0f;
}

// B-matrix [N rows, K cols] row-major -> K-pair-interleaved:
// out[((k>>1)*N + n)*2 + (k&1)] = in[n*K + k]
__global__ void repack_nk_interleave(const float* __restrict__ in, float* __restrict__ out,
                                     int N, int K) {
  long ntot = (long)N * K;
  long o = (long)blockIdx.x * blockDim.x + threadIdx.x;
  long stride = (long)gridDim.x * blockDim.x;
  for (; o < ntot; o += stride) {
    int q = (int)(o & 1);
    long t = o >> 1;
    int n = (int)(t % N);
    int k = (int)(2 * (t / N) + q);
    out[o] = in[(long)n * K + k];
  }
}

// conv weight [E][Cin][3][3]; logical B[k][n] with k = tap*Cin + c, n = e
// -> K-pair-interleaved layout as above (N = E, K = 9*Cin)
__global__ void repack_conv_interleave(const float* __restrict__ w, float* __restrict__ out,
                                       int Cin, int E) {
  long ntot = (long)E * Cin * 9;
  long o = (long)blockIdx.x * blockDim.x + threadIdx.x;
  long stride = (long)gridDim.x * blockDim.x;
  for (; o < ntot; o += stride) {
    int q = (int)(o & 1);
    long t = o >> 1;
    int n = (int)(t % E);
    int k = (int)(2 * (t / E) + q);
    int tap = k / Cin, c = k % Cin;
    out[o] = w[((long)n * Cin + c) * 9 + tap];
  }
}

// P[b*NVALID+v][c] = max_{n in [i..j]} AV[b][n][c]  (pool chain == span max)
__global__ void pool_kernel(const float* __restrict__ AV, float* __restrict__ P) {
  int bid = blockIdx.x;
  int v = bid % NVALID;
  int b = bid / NVALID;
  int c = threadIdx.x;
  int i = g_tab.vi[v], j = g_tab.vj[v];
  const float* base = AV + (long)b * NSEG * DMODEL + c;
  float mx = base[(long)i * DMODEL];
  for (int n = i + 1; n <= j; ++n) mx = fmaxf(mx, base[(long)n * DMODEL]);
  P[(long)bid * DMODEL + c] = mx;
}

// channel-wise L2 normalize Y rows and scatter into padded H0 grid [B,68,68,256]
__global__ void norm_scatter_kernel(const float* __restrict__ Y, float* __restrict__ H0) {
  __shared__ float s[DMODEL];
  int bid = blockIdx.x;
  int v = bid % NVALID;
  int b = bid / NVALID;
  int c = threadIdx.x;
  float y = Y[(long)bid * DMODEL + c];
  s[c] = y * y;
  __syncthreads();
  for (int off = DMODEL / 2; off > 0; off >>= 1) {
    if (c < off) s[c] += s[c + off];
    __syncthreads();
  }
  float scale = 1.0f / fmaxf(sqrtf(s[0]), 1e-12f);
  int i = g_tab.vi[v], j = g_tab.vj[v];
  H0[(((long)b * 68 + (i + 2)) * 68 + (j + 2)) * DMODEL + c] = y * scale;
}

// ---------------- WMMA fp32 GEMM: Out[M,N] = A[M,K] * B[K,N] + bias ----------------
// Wt2 = K-pair-interleaved B. One wave computes 16 rows x 64 cols (NT=4 tiles).
// M%16==0, N%64==0, K%4==0.
__global__ void gemm_wmma_kernel(const float* __restrict__ A, const float* __restrict__ Wt2,
                                 const float* __restrict__ bias, float* __restrict__ Out,
                                 int K, int N) {
  int lane = threadIdx.x;
  int row = lane & 15;
  int kh = lane >> 4;
  long m = (long)blockIdx.x * 16 + row;
  int nbase = blockIdx.y * (16 * NT);
  const float* arow = A + m * (long)K + 2 * kh;
  // pair index p = k0/2 + kh ; float addr = (p*N + col)*2
  const float* bptr = Wt2 + ((long)kh * N + nbase + row) * 2;
  v8f acc[NT] = {};
  for (int k0 = 0; k0 < K; k0 += 4) {
    v2f a = *(const v2f*)arow;
#pragma unroll
    for (int nt = 0; nt < NT; ++nt) {
      v2f bb = *(const v2f*)(bptr + nt * 32);   // +16 cols = 32 floats
      acc[nt] = __builtin_amdgcn_wmma_f32_16x16x4_f32(false, a, false, bb, (short)0,
                                                      acc[nt], false, false);
    }
    arow += 4;
    bptr += (long)4 * N;  // 2 pair-rows forward
  }
  long mbase = (long)blockIdx.x * 16 + 8 * kh;
#pragma unroll
  for (int nt = 0; nt < NT; ++nt) {
    int ncol = nbase + nt * 16 + row;
    float bi = bias[ncol];
#pragma unroll
    for (int r = 0; r < 8; ++r)
      Out[(mbase + r) * (long)N + ncol] = acc[nt][r] + bi;
  }
}

// ---------------- WMMA 3x3 conv (implicit GEMM), relu * maskweight epilogue ----------
// act: [B, outH+2, outW+2, Cin]; wt2: K-pair-interleaved (K = 9*Cin, N = Cout)
// valid_mode=0: out full grid [B,outH,outW,Cout];  valid_mode=1: out rows [B*NVALID,Cout]
__global__ void conv3x3_wmma_kernel(const float* __restrict__ act, int Cin,
                                    const float* __restrict__ wt2,
                                    const float* __restrict__ bias,
                                    float* __restrict__ out,
                                    int outH, int outW, int Cout,
                                    int mw_sel, int valid_mode) {
  int lane = threadIdx.x;
  int row = lane & 15, kh = lane >> 4;
  int actW = outW + 2, actH = outH + 2;
  long m = (long)blockIdx.x * 16 + row;
  int nbase = blockIdx.y * (16 * NT);
  int b, oy, ox;
  if (valid_mode) {
    b = (int)(m / NVALID);
    int v = (int)(m % NVALID);
    oy = g_tab.vi[v]; ox = g_tab.vj[v];
  } else {
    int hw = outH * outW;
    b = (int)(m / hw);
    int p = (int)(m % hw);
    oy = p / outW; ox = p % outW;
  }
  const float* actb = act + (long)b * actH * actW * Cin;
  v8f acc[NT] = {};
  for (int ky = 0; ky < 3; ++ky)
    for (int kx = 0; kx < 3; ++kx) {
      int tap = ky * 3 + kx;
      const float* arow = actb + ((long)(oy + ky) * actW + (ox + kx)) * Cin + 2 * kh;
      // pair index p = tap*Cin/2 + c0/2 + kh
      const float* bptr = wt2 + (((long)tap * (Cin / 2) + kh) * Cout + nbase + row) * 2;
      for (int c0 = 0; c0 < Cin; c0 += 4) {
        v2f a = *(const v2f*)arow;
#pragma unroll
        for (int nt = 0; nt < NT; ++nt) {
          v2f bb = *(const v2f*)(bptr + nt * 32);
          acc[nt] = __builtin_amdgcn_wmma_f32_16x16x4_f32(false, a, false, bb, (short)0,
                                                          acc[nt], false, false);
        }
        arow += 4;
        bptr += (long)4 * Cout;
      }
    }
#pragma unroll
  for (int r = 0; r < 8; ++r) {
    long mm = (long)blockIdx.x * 16 + 8 * kh + r;
    int oy2, ox2;
    long obase;
    if (valid_mode) {
      int v2_ = (int)(mm % NVALID);
      oy2 = g_tab.vi[v2_]; ox2 = g_tab.vj[v2_];
      obase = mm * (long)Cout;
    } else {
      int hw = outH * outW;
      int b2 = (int)(mm / hw);
      int p = (int)(mm % hw);
      oy2 = p / outW; ox2 = p % outW;
      obase = (((long)b2 * outH + oy2) * outW + ox2) * (long)Cout;
    }
    float w = (mw_sel == 0) ? g_tab.mw0[oy2 * 66 + ox2] : g_tab.mw1[oy2 * 64 + ox2];
#pragma unroll
    for (int nt = 0; nt < NT; ++nt) {
      int ncol = nbase + nt * 16 + row;
      out[obase + ncol] = fmaxf(acc[nt][r] + bias[ncol], 0.0f) * w;
    }
  }
}

// ---------------- launch ----------------
extern "C" void kernel_launch(void* const* d_in, const int* in_sizes, int n_in,
                              void* d_out, int out_size, void* d_ws, size_t ws_size,
                              hipStream_t stream) {
  const float* AV   = (const float*)d_in[0];
  // d_in[1] = S (unused by the reference)
  const float* w1x1 = (const float*)d_in[2];
  const float* b1x1 = (const float*)d_in[3];
  const float* cw0  = (const float*)d_in[4];
  const float* cb0  = (const float*)d_in[5];
  const float* cw1  = (const float*)d_in[6];
  const float* cb1  = (const float*)d_in[7];
  const float* fcw  = (const float*)d_in[8];
  const float* fcb  = (const float*)d_in[9];
  float* out = (float*)d_out;
  float* ws  = (float*)d_ws;

  size_t off = 0;
  float* w1x1t = ws + off; off += (size_t)DMODEL * DMODEL;        // interleaved [256x256]
  float* fcwt  = ws + off; off += (size_t)HDIM * DMODEL;          // interleaved [512x256]
  float* wt0   = ws + off; off += (size_t)9 * DMODEL * HDIM;      // interleaved [2304x512]
  float* wt1   = ws + off; off += (size_t)9 * HDIM * HDIM;        // interleaved [4608x512]
  float* P     = ws + off; off += (size_t)BATCH * NVALID * DMODEL;
  float* Y     = ws + off; off += (size_t)BATCH * NVALID * DMODEL;
  float* H0    = ws + off; off += (size_t)BATCH * 68 * 68 * DMODEL;
  float* H1    = ws + off; off += (size_t)BATCH * 66 * 66 * HDIM;
  float* H2    = ws + off; off += (size_t)BATCH * NVALID * HDIM;

  // 1) zero padded H0 grid (invalid + pad cells must be 0)
  zero_kernel<<<2048, 256, 0, stream>>>(H0, (long)BATCH * 68 * 68 * DMODEL);
  // 2) weight repacks into K-pair-interleaved B layout
  repack_nk_interleave<<<256, 256, 0, stream>>>(w1x1, w1x1t, DMODEL, DMODEL);
  repack_nk_interleave<<<512, 256, 0, stream>>>(fcw, fcwt, DMODEL, HDIM);
  repack_conv_interleave<<<2048, 256, 0, stream>>>(cw0, wt0, DMODEL, HDIM);
  repack_conv_interleave<<<4096, 256, 0, stream>>>(cw1, wt1, HDIM, HDIM);
  // 3) multi-scale pooling == span max at the 1104 valid cells
  pool_kernel<<<BATCH * NVALID, DMODEL, 0, stream>>>(AV, P);
  // 4) 1x1 conv as GEMM [4416,256]x[256,256]
  gemm_wmma_kernel<<<dim3(BATCH * NVALID / 16, DMODEL / (16 * NT)), 32, 0, stream>>>(
      P, w1x1t, b1x1, Y, DMODEL, DMODEL);
  // 5) L2-normalize per position, scatter into padded grid
  norm_scatter_kernel<<<BATCH * NVALID, DMODEL, 0, stream>>>(Y, H0);
  // 6) conv0: 3x3, 256->512, pad2 -> 66x66, relu*MW0
  conv3x3_wmma_kernel<<<dim3(BATCH * 66 * 66 / 16, HDIM / (16 * NT)), 32, 0, stream>>>(
      H0, DMODEL, wt0, cb0, H1, 66, 66, HDIM, /*mw*/0, /*valid*/0);
  // 7) conv1: 3x3, 512->512, valid cells only, relu*MW1 -> rows [4416,512]
  conv3x3_wmma_kernel<<<dim3(BATCH * NVALID / 16, HDIM / (16 * NT)), 32, 0, stream>>>(
      H1, HDIM, wt1, cb1, H2, 64, 64, HDIM, /*mw*/1, /*valid*/1);
  // 8) fc: [4416,512]x[512,256] -> d_out (bs,ns,1104,256)
  gemm_wmma_kernel<<<dim3(BATCH * NVALID / 16, DMODEL / (16 * NT)), 32, 0, stream>>>(
      H2, fcwt, fcb, out, HDIM, DMODEL);
}